// MuyGP_84482006712737
// MI455X (gfx1250) — compile-verified
//
#include <hip/hip_runtime.h>
#include <math.h>

typedef __attribute__((ext_vector_type(2))) float v2f;
typedef __attribute__((ext_vector_type(8))) float v8f;

#define DD 64          // feature dim
#define NNB 128        // neighborhood size
#define STRIP 8192     // train points scored per strip
#define QTILE 16       // queries per score block
#define PTB 128        // points per score block (8 waves * 16)
#define INF_F 3.402823466e38f

// ---------------------------------------------------------------- norms
__global__ __launch_bounds__(256) void k_norms(const float* __restrict__ trainX,
                                               float* __restrict__ tnorm, int N) {
    int i = blockIdx.x * blockDim.x + threadIdx.x;
    if (i >= N) return;
    const float* r = trainX + (size_t)i * DD;
    float s = 0.f;
#pragma unroll
    for (int k = 0; k < DD; ++k) s += r[k] * r[k];
    tnorm[i] = s;
}

// ---------------------------------------------------------------- init best lists
__global__ __launch_bounds__(NNB) void k_init_best(float* __restrict__ bs,
                                                   int* __restrict__ bi) {
    size_t o = (size_t)blockIdx.x * NNB + threadIdx.x;
    bs[o] = INF_F;
    bi[o] = 0;
}

// ---------------------------------------------------------------- score GEMM (f32 WMMA)
// score[q][p] = |t_p|^2 - 2 * x_q . t_p   (|x_q|^2 constant per row; ranking-equivalent)
__global__ __launch_bounds__(256) void k_score(const float* __restrict__ x,
                                               const float* __restrict__ trainX,
                                               const float* __restrict__ tnorm,
                                               float* __restrict__ scores,
                                               int N, int stripBase) {
    const int lane = threadIdx.x & 31;
    const int wave = threadIdx.x >> 5;
    const int half = lane >> 4;   // 0: lanes 0-15, 1: lanes 16-31
    const int l15  = lane & 15;
    const int qBase   = blockIdx.x * QTILE;
    const int pLocal0 = blockIdx.y * PTB + wave * 16;

    // A: 16 queries x 64, as 16 chained 16x4 f32 fragments
    v2f afrag[16];
    {
        const float* xr = x + (size_t)(qBase + l15) * DD;
#pragma unroll
        for (int k = 0; k < 16; ++k) {
            int kb = 4 * k + 2 * half;
            afrag[k].x = xr[kb];
            afrag[k].y = xr[kb + 1];
        }
    }

    // B: 16 train points x 64 (row-clamped at the tail strip; merge kernel guards)
    int gp  = stripBase + pLocal0 + l15;
    int gpc = gp < N ? gp : (N - 1);
    const float* tr = trainX + (size_t)gpc * DD;

    v8f c = {};
#pragma unroll
    for (int k = 0; k < 16; ++k) {
        int kb = 4 * k + 2 * half;
        v2f b;
        b.x = tr[kb];
        b.y = tr[kb + 1];
        c = __builtin_amdgcn_wmma_f32_16x16x4_f32(false, afrag[k], false, b,
                                                  (short)0, c, false, false);
    }

    // C/D layout: col = lane&15 ; row = r + 8*(lane>=16)
    int pl = pLocal0 + l15;
    int g  = stripBase + pl;
    float nrm = tnorm[g < N ? g : (N - 1)];
#pragma unroll
    for (int r = 0; r < 8; ++r) {
        int q = qBase + r + 8 * half;
        scores[(size_t)q * STRIP + pl] = nrm - 2.0f * c[r];
    }
}

// ---------------------------------------------------------------- running top-128 merge
__global__ __launch_bounds__(NNB) void k_topk(const float* __restrict__ scores,
                                              float* __restrict__ bs,
                                              int* __restrict__ bi,
                                              int N, int stripBase) {
    __shared__ float sBest[NNB];
    __shared__ int   iBest[NNB];
    __shared__ float sChk[NNB];
    __shared__ int   iChk[NNB];
    const int t = threadIdx.x;
    const int q = blockIdx.x;

    sBest[t] = bs[(size_t)q * NNB + t];
    iBest[t] = bi[(size_t)q * NNB + t];
    __syncthreads();

    for (int c0 = 0; c0 < STRIP; c0 += NNB) {
        int   pl = c0 + t;
        int   g  = stripBase + pl;
        float s  = (g < N) ? scores[(size_t)q * STRIP + pl] : INF_F;
        float kth = sBest[NNB - 1];
        int cnt = __syncthreads_count(s < kth);   // uniform; barrier included
        if (cnt == 0) continue;                   // fast path after warm-up

        sChk[t] = s;
        iChk[t] = g;
        __syncthreads();

        // bitonic sort chunk ascending (128 keys, 128 threads)
        for (int k = 2; k <= NNB; k <<= 1) {
            for (int j = k >> 1; j > 0; j >>= 1) {
                int ixj = t ^ j;
                if (ixj > t) {
                    bool up = ((t & k) == 0);
                    float a = sChk[t], b = sChk[ixj];
                    if (up ? (a > b) : (a < b)) {
                        sChk[t] = b; sChk[ixj] = a;
                        int ia = iChk[t]; iChk[t] = iChk[ixj]; iChk[ixj] = ia;
                    }
                }
                __syncthreads();
            }
        }

        // keep 128 smallest of best ∪ chunk: min(A[t], B[127-t]) is bitonic
        float ma = sBest[t], mb = sChk[NNB - 1 - t];
        float ms; int mi;
        if (ma <= mb) { ms = ma; mi = iBest[t]; }
        else          { ms = mb; mi = iChk[NNB - 1 - t]; }
        __syncthreads();
        sBest[t] = ms;
        iBest[t] = mi;
        __syncthreads();

        // clean the bitonic sequence -> ascending (7 stages)
        for (int j = NNB >> 1; j > 0; j >>= 1) {
            int ixj = t ^ j;
            if (ixj > t) {
                float av = sBest[t], bv = sBest[ixj];
                if (av > bv) {
                    sBest[t] = bv; sBest[ixj] = av;
                    int ia = iBest[t]; iBest[t] = iBest[ixj]; iBest[ixj] = ia;
                }
            }
            __syncthreads();
        }
    }

    bs[(size_t)q * NNB + t] = sBest[t];
    bi[(size_t)q * NNB + t] = iBest[t];
}

// ---------------------------------------------------------------- per-query solve
// Gram(WMMA from LDS) -> RBF -> Cholesky -> two tri-solves -> y, yVar
__global__ __launch_bounds__(256) void k_solve(const float* __restrict__ x,
                                               const float* __restrict__ trainX,
                                               const float* __restrict__ trainy,
                                               const int* __restrict__ bi,
                                               const float* __restrict__ lP,
                                               const float* __restrict__ aP,
                                               float* __restrict__ out, int Bq) {
    extern __shared__ float lds[];
    float* NXs = lds;                      // 128*64
    float* K   = NXs + NNB * DD;           // 128*129 (padded: conflict-free columns)
    float* Xq  = K + NNB * (NNB + 1);      // 64
    float* nrm = Xq + DD;                  // 128
    float* cv  = nrm + NNB;                // 128 working copy of crossCov
    float* c0  = cv + NNB;                 // 128 pristine crossCov
    float* zw  = c0 + NNB;                 // 128 z (fwd) / residual (bwd)
    float* nyv = zw + NNB;                 // 128 gathered trainy
    float* red = nyv + NNB;                // 2*128 reduction

    const int t = threadIdx.x;             // 0..255
    const int q = blockIdx.x;
    const float l = *lP, a = *aP;
    const float inv2l2 = 1.0f / (2.0f * l * l);

    for (int k = t; k < DD; k += blockDim.x) Xq[k] = x[(size_t)q * DD + k];
    for (int e = t; e < NNB * DD; e += blockDim.x) {
        int i = e >> 6, k = e & 63;
        NXs[e] = trainX[(size_t)bi[(size_t)q * NNB + i] * DD + k];
    }
    if (t < NNB) nyv[t] = trainy[bi[(size_t)q * NNB + t]];
    __syncthreads();

    if (t < NNB) {
        float s = 0.f;
        const float* r = NXs + t * DD;
#pragma unroll
        for (int k = 0; k < DD; ++k) s += r[k] * r[k];
        nrm[t] = s;
    }
    __syncthreads();

    // 64 tiles of 16x16; 8 waves * 8 tiles; f32 WMMA chain over K=64, A/B from LDS
    const int lane = t & 31, wave = t >> 5;
    const int half = lane >> 4, l15 = lane & 15;
    for (int tt = wave * 8; tt < wave * 8 + 8; ++tt) {
        int tm = tt >> 3, tn = tt & 7;
        const float* ar = NXs + (tm * 16 + l15) * DD;
        const float* br = NXs + (tn * 16 + l15) * DD;
        v8f cacc = {};
#pragma unroll
        for (int k = 0; k < 16; ++k) {
            int kb = 4 * k + 2 * half;
            v2f av; av.x = ar[kb]; av.y = ar[kb + 1];
            v2f bv; bv.x = br[kb]; bv.y = br[kb + 1];
            cacc = __builtin_amdgcn_wmma_f32_16x16x4_f32(false, av, false, bv,
                                                         (short)0, cacc, false, false);
        }
        int col = tn * 16 + l15;
#pragma unroll
        for (int r = 0; r < 8; ++r) {
            int row = tm * 16 + r + 8 * half;
            float sq = nrm[row] + nrm[col] - 2.0f * cacc[r];
            sq = sq > 0.f ? sq : 0.f;
            K[row * (NNB + 1) + col] = a * expf(-sq * inv2l2);
        }
    }
    __syncthreads();

    // in-place Cholesky (lower), left-looking
    for (int j = 0; j < NNB; ++j) {
        if (t == j) {
            float s = K[j * (NNB + 1) + j];
            for (int p = 0; p < j; ++p) { float v = K[j * (NNB + 1) + p]; s -= v * v; }
            K[j * (NNB + 1) + j] = sqrtf(fmaxf(s, 1e-20f));
        }
        __syncthreads();
        if (t > j && t < NNB) {
            float s = K[t * (NNB + 1) + j];
            for (int p = 0; p < j; ++p) s -= K[t * (NNB + 1) + p] * K[j * (NNB + 1) + p];
            K[t * (NNB + 1) + j] = s / K[j * (NNB + 1) + j];
        }
        __syncthreads();
    }

    // crossCov
    if (t < NNB) {
        float dt = 0.f, qq = 0.f;
        const float* r = NXs + t * DD;
#pragma unroll
        for (int k = 0; k < DD; ++k) { dt += Xq[k] * r[k]; qq += Xq[k] * Xq[k]; }
        float sq = fmaxf(qq + nrm[t] - 2.f * dt, 0.f);
        float v = a * expf(-sq * inv2l2);
        cv[t] = v; c0[t] = v;
    }
    __syncthreads();

    // L z = c (forward)
    for (int j = 0; j < NNB; ++j) {
        if (t == j) zw[j] = cv[j] / K[j * (NNB + 1) + j];
        __syncthreads();
        if (t > j && t < NNB) cv[t] -= K[t * (NNB + 1) + j] * zw[j];
        __syncthreads();
    }
    // L^T w = z (backward), w lands in cv
    for (int j = NNB - 1; j >= 0; --j) {
        if (t == j) cv[j] = zw[j] / K[j * (NNB + 1) + j];
        __syncthreads();
        if (t < j) zw[t] -= K[j * (NNB + 1) + t] * cv[j];
        __syncthreads();
    }

    if (t < NNB) {
        red[t]       = cv[t] * nyv[t];
        red[NNB + t] = cv[t] * c0[t];
    }
    __syncthreads();
    for (int s = NNB >> 1; s > 0; s >>= 1) {
        if (t < s) { red[t] += red[t + s]; red[NNB + t] += red[NNB + t + s]; }
        __syncthreads();
    }
    if (t == 0) {
        out[q]      = red[0];
        out[Bq + q] = a - red[NNB];
    }
}

// ---------------------------------------------------------------- host
extern "C" void kernel_launch(void* const* d_in, const int* in_sizes, int n_in,
                              void* d_out, int out_size, void* d_ws, size_t ws_size,
                              hipStream_t stream) {
    const float* x      = (const float*)d_in[0];
    const float* trainX = (const float*)d_in[1];
    const float* trainy = (const float*)d_in[2];
    const float* lP     = (const float*)d_in[3];
    const float* aP     = (const float*)d_in[4];
    float* out = (float*)d_out;

    const int Bq = in_sizes[0] / DD;   // 2048
    const int N  = in_sizes[1] / DD;   // 100000

    // workspace partition (256B aligned):
    // tnorm[N] | bestScore[Bq*128] | bestIdx[Bq*128] | scores[Bq*STRIP]
    char* w = (char*)d_ws;
    size_t o = 0;
    float* tnorm  = (float*)(w + o); o = (o + (size_t)N * 4 + 255) & ~(size_t)255;
    float* bs     = (float*)(w + o); o = (o + (size_t)Bq * NNB * 4 + 255) & ~(size_t)255;
    int*   bi     = (int*)  (w + o); o = (o + (size_t)Bq * NNB * 4 + 255) & ~(size_t)255;
    float* scores = (float*)(w + o);
    (void)ws_size; (void)n_in; (void)out_size;

    k_norms<<<dim3((N + 255) / 256), dim3(256), 0, stream>>>(trainX, tnorm, N);
    k_init_best<<<dim3(Bq), dim3(NNB), 0, stream>>>(bs, bi);

    for (int sb = 0; sb < N; sb += STRIP) {
        k_score<<<dim3(Bq / QTILE, STRIP / PTB), dim3(256), 0, stream>>>(
            x, trainX, tnorm, scores, N, sb);
        k_topk<<<dim3(Bq), dim3(NNB), 0, stream>>>(scores, bs, bi, N, sb);
    }

    const size_t shmem = (size_t)(NNB * DD + NNB * (NNB + 1) + DD + 7 * NNB) * sizeof(float);
    k_solve<<<dim3(Bq), dim3(256), shmem, stream>>>(x, trainX, trainy, bi, lP, aP, out, Bq);
}